// MultiBoxLoss_90271622627423
// MI455X (gfx1250) — compile-verified
//
#include <hip/hip_runtime.h>
#include <hip/hip_bf16.h>

// ---------------- problem constants ----------------
#define NB   64
#define NP   8732
#define NOBJ 16
#define NC   81
#define TILES_PER_B   546            // ceil(8732/16)
#define TOTAL_WAVES   (NB * TILES_PER_B)   // 34944
#define SORT_N 16384

typedef __attribute__((ext_vector_type(2))) float v2f;
typedef __attribute__((ext_vector_type(8))) float v8f;

__device__ __forceinline__ float smooth_l1(float x) {
    float ax = fabsf(x);
    return (ax < 1.0f) ? 0.5f * x * x : ax - 0.5f;
}

// ---------------- kernel 0: zero accumulators ----------------
__global__ void mb_init_kernel(float* accf, int* npos_tot, int* npos) {
    int t = threadIdx.x;
    if (t < 4) accf[t] = 0.0f;           // [0]=loss_l  [1]=sum_pos_lc  [2]=topk  [3]=pad
    if (t == 0) *npos_tot = 0;
    if (t < NB) npos[t] = 0;
}

// ---------------- kernel 1: matching (one block per batch) ----------------
__global__ void mb_match_kernel(const float* __restrict__ targets,
                                const float* __restrict__ priors,
                                float* __restrict__ bto, int* __restrict__ bti) {
    const int b   = blockIdx.x;
    const int tid = threadIdx.x;
    __shared__ float tr[NOBJ][4];
    __shared__ unsigned long long best[NOBJ];   // packed (overlap_bits<<32) | ~prior

    if (tid < NOBJ * 4) tr[tid >> 2][tid & 3] = targets[(size_t)b * NOBJ * 5 + (tid >> 2) * 5 + (tid & 3)];
    if (tid < NOBJ) best[tid] = 0ULL;
    __syncthreads();

    float tbo[NOBJ];
    unsigned tbp[NOBJ];
#pragma unroll
    for (int t = 0; t < NOBJ; ++t) { tbo[t] = -1.0f; tbp[t] = 0; }

    for (int p = tid; p < NP; p += blockDim.x) {
        float cx = priors[p * 4 + 0], cy = priors[p * 4 + 1];
        float w  = priors[p * 4 + 2], h  = priors[p * 4 + 3];
        float px1 = cx - 0.5f * w, py1 = cy - 0.5f * h;
        float px2 = cx + 0.5f * w, py2 = cy + 0.5f * h;
        float areap = w * h;
        float bo = -1.0f; int bidx = 0;
#pragma unroll
        for (int t = 0; t < NOBJ; ++t) {
            float x1 = tr[t][0], y1 = tr[t][1], x2 = tr[t][2], y2 = tr[t][3];
            float iw = fmaxf(fminf(px2, x2) - fmaxf(px1, x1), 0.0f);
            float ih = fmaxf(fminf(py2, y2) - fmaxf(py1, y1), 0.0f);
            float inter = iw * ih;
            float at = (x2 - x1) * (y2 - y1);
            float ov = inter / (at + areap - inter);
            if (ov > bo) { bo = ov; bidx = t; }
            if (ov > tbo[t]) { tbo[t] = ov; tbp[t] = (unsigned)p; }
        }
        bto[(size_t)b * NP + p] = bo;
        bti[(size_t)b * NP + p] = bidx;
    }
#pragma unroll
    for (int t = 0; t < NOBJ; ++t) {
        float ov = fmaxf(tbo[t], 0.0f);
        unsigned long long key = ((unsigned long long)__float_as_uint(ov) << 32)
                               | (unsigned long long)(0xFFFFFFFFu - tbp[t]);
        atomicMax(&best[t], key);
    }
    __threadfence();
    __syncthreads();
    if (tid == 0) {   // in-order overrides: later truth wins on collision
        for (int j = 0; j < NOBJ; ++j) {
            unsigned bp = 0xFFFFFFFFu - (unsigned)(best[j] & 0xFFFFFFFFu);
            bto[(size_t)b * NP + bp] = 2.0f;
            bti[(size_t)b * NP + bp] = j;
        }
    }
}

// ---------------- kernel 2: fused LSE (via WMMA) + conf/loc losses ----------------
// One wave32 per 16-prior tile. A(16x4) = exp slab, B = ones(4x16):
// D columns = per-prior sum(exp) over the 81 classes, accumulated over 21 K-steps.
// Branch-free load path: invalid lanes read a clamped (legal) row and are
// zeroed in ALU so the compiler can batch all global loads ahead of the
// exp/wmma chain (one wait instead of 21).
__global__ void mb_loss_kernel(const float* __restrict__ loc,
                               const float* __restrict__ conf,
                               const float* __restrict__ priors,
                               const float* __restrict__ targets,
                               const float* __restrict__ bto,
                               const int*   __restrict__ bti,
                               float* __restrict__ lcm,
                               float* __restrict__ accf,
                               int* __restrict__ npos, int* __restrict__ npos_tot) {
    const int wave = (blockIdx.x * blockDim.x + threadIdx.x) >> 5;
    const int lane = threadIdx.x & 31;
    if (wave >= TOTAL_WAVES) return;
    const int b     = wave / TILES_PER_B;
    const int tbase = (wave % TILES_PER_B) * 16;
    const int row   = lane & 15;
    const int p     = tbase + row;
    const bool valid = (p < NP);
    const int  pc    = valid ? p : (NP - 1);          // clamped, always legal
    const float m    = valid ? 1.0f : 0.0f;           // ALU mask, no divergence

    const float* crow = conf + ((size_t)b * NP + pc) * NC;
    const int cbase = (lane < 16) ? 0 : 2;   // 16x4 f32 A layout: lanes0-15 -> K0,K1 ; lanes16-31 -> K2,K3

    // ---- gather all 40 in-bounds operands (cols 0..79) + col 80 up front ----
    float ev[40];
#pragma unroll
    for (int t = 0; t < 20; ++t) {
        ev[2 * t + 0] = crow[t * 4 + cbase + 0];
        ev[2 * t + 1] = crow[t * 4 + cbase + 1];
    }
    float v80 = crow[80];                    // in-bounds for every row, all lanes

    v2f ones; ones.x = 1.0f; ones.y = 1.0f;
    v8f acc = {};
#pragma unroll
    for (int t = 0; t < 20; ++t) {
        v2f a;
        a.x = m * __expf(ev[2 * t + 0]);
        a.y = m * __expf(ev[2 * t + 1]);
        acc = __builtin_amdgcn_wmma_f32_16x16x4_f32(
                  false, a, false, ones, (short)0, acc, false, false);
    }
    {   // final K-step: only column 80 exists; lanes >= 16 (K=2,3) contribute 0
        v2f a;
        a.x = (lane < 16) ? (m * __expf(v80)) : 0.0f;
        a.y = 0.0f;
        acc = __builtin_amdgcn_wmma_f32_16x16x4_f32(
                  false, a, false, ones, (short)0, acc, false, false);
    }

    // D layout: lane0 holds rows 0..7 (N=0) in acc[0..7]; lane16 holds rows 8..15.
    float rs = 0.0f;
#pragma unroll
    for (int v = 0; v < 8; ++v) {
        float lo = __shfl(acc[v], 0, 32);
        float hi = __shfl(acc[v], 16, 32);
        if (lane == v)     rs = lo;
        if (lane == v + 8) rs = hi;
    }

    bool posFlag = false;
    if (lane < 16 && valid) {
        const size_t idx = (size_t)b * NP + p;
        const int   ti = bti[idx];
        const float ov = bto[idx];
        const float* tg = targets + ((size_t)b * NOBJ + ti) * 5;
        float x1 = tg[0], y1 = tg[1], x2 = tg[2], y2 = tg[3];
        int ct = (ov < 0.5f) ? 0 : ((int)tg[4] + 1);
        float lse = __logf(rs);
        float lc  = lse - crow[ct];
        posFlag   = (ct > 0);
        lcm[idx]  = posFlag ? 0.0f : lc;
        if (posFlag) {
            atomicAdd(&accf[1], lc);
            float pcx = priors[p * 4 + 0], pcy = priors[p * 4 + 1];
            float pw  = priors[p * 4 + 2], ph  = priors[p * 4 + 3];
            float t0 = (0.5f * (x1 + x2) - pcx) / (0.1f * pw);
            float t1 = (0.5f * (y1 + y2) - pcy) / (0.1f * ph);
            float t2 = __logf((x2 - x1) / pw) * 5.0f;   // /0.2
            float t3 = __logf((y2 - y1) / ph) * 5.0f;
            const float* lrow = loc + idx * 4;
            float s = smooth_l1(lrow[0] - t0) + smooth_l1(lrow[1] - t1)
                    + smooth_l1(lrow[2] - t2) + smooth_l1(lrow[3] - t3);
            atomicAdd(&accf[0], s);
        }
    }
    unsigned long long bal = __ballot(posFlag);
    if (lane == 0 && bal) {
        int cnt = __popcll(bal);
        atomicAdd(&npos[b], cnt);
        atomicAdd(npos_tot, cnt);
    }
}

// ---------------- kernel 3: per-batch top-k sum via LDS bitonic sort ----------------
__global__ void mb_topk_kernel(const float* __restrict__ lcm,
                               const int* __restrict__ npos,
                               float* __restrict__ accf) {
    const int b = blockIdx.x, tid = threadIdx.x, nt = blockDim.x;
    __shared__ float s[SORT_N];
    for (int i = tid; i < SORT_N; i += nt)
        s[i] = (i < NP) ? lcm[(size_t)b * NP + i] : -1.0f;   // loss_c >= 0, pad never selected
    __syncthreads();

    for (unsigned k = 2; k <= SORT_N; k <<= 1) {
        for (unsigned j = k >> 1; j > 0; j >>= 1) {
            for (unsigned i = tid; i < SORT_N; i += nt) {
                unsigned ixj = i ^ j;
                if (ixj > i) {
                    float a = s[i], bb = s[ixj];
                    bool up = ((i & k) == 0);
                    if (up ? (a < bb) : (a > bb)) { s[i] = bb; s[ixj] = a; }
                }
            }
            __syncthreads();
        }
    }
    int nn = min(3 * npos[b], NP - 1);
    float part = 0.0f;
    for (int i = tid; i < nn; i += nt) part += s[i];
    __syncthreads();
    s[tid] = part;
    __syncthreads();
    for (int st = nt >> 1; st > 0; st >>= 1) {
        if (tid < st) s[tid] += s[tid + st];
        __syncthreads();
    }
    if (tid == 0) atomicAdd(&accf[2], s[0]);
}

// ---------------- kernel 4: finalize ----------------
__global__ void mb_final_kernel(const float* __restrict__ accf,
                                const int* __restrict__ npos_tot,
                                float* __restrict__ out) {
    if (threadIdx.x == 0 && blockIdx.x == 0) {
        float N = (float)max(*npos_tot, 1);
        out[0] = (accf[0] + accf[1] + accf[2]) / N;
    }
}

extern "C" void kernel_launch(void* const* d_in, const int* in_sizes, int n_in,
                              void* d_out, int out_size, void* d_ws, size_t ws_size,
                              hipStream_t stream) {
    const float* loc     = (const float*)d_in[0];   // (B,P,4)
    const float* conf    = (const float*)d_in[1];   // (B,P,81)
    const float* priors  = (const float*)d_in[2];   // (P,4)
    const float* targets = (const float*)d_in[3];   // (B,16,5)

    const size_t BP = (size_t)NB * NP;
    float* bto  = (float*)d_ws;          // B*P f32
    int*   bti  = (int*)(bto + BP);      // B*P i32
    float* lcm  = (float*)(bti + BP);    // B*P f32 (loss_c_mine)
    float* accf = lcm + BP;              // [loss_l, sum_pos_lc, topk, pad]
    int*   ip   = (int*)(accf + 4);
    int*   npos_tot = ip;                // 1
    int*   npos     = ip + 1;            // B

    mb_init_kernel<<<1, 128, 0, stream>>>(accf, npos_tot, npos);
    mb_match_kernel<<<NB, 256, 0, stream>>>(targets, priors, bto, bti);
    const int blocks = (TOTAL_WAVES * 32 + 255) / 256;   // 8 waves/block
    mb_loss_kernel<<<blocks, 256, 0, stream>>>(loc, conf, priors, targets,
                                               bto, bti, lcm, accf, npos, npos_tot);
    mb_topk_kernel<<<NB, 256, 0, stream>>>(lcm, npos, accf);
    mb_final_kernel<<<1, 32, 0, stream>>>(accf, npos_tot, (float*)d_out);
}